// GNN_15006615734387
// MI455X (gfx1250) — compile-verified
//
#include <hip/hip_runtime.h>
#include <hip/hip_bf16.h>

typedef float v2f __attribute__((ext_vector_type(2)));
typedef float v8f __attribute__((ext_vector_type(8)));

#define GN_N 50000
#define GN_E 800000
#define GN_B 256
#define GN_D 128
#define GN_L 4
#define ASTRIDE 388   // 384 + 4 pad: breaks 64-bank conflict (384 % 64 == 0)

// ---------------- utility ----------------
__global__ __launch_bounds__(256) void fillz(float* __restrict__ p, int n) {
    int i = blockIdx.x * 256 + threadIdx.x;
    if (i < n) p[i] = 0.0f;
}

// ---------------- degree / d^-1/2 ----------------
__global__ __launch_bounds__(256) void deg_scatter(const int* __restrict__ dst,
                                                   float* __restrict__ deg, int E) {
    int e = blockIdx.x * 256 + threadIdx.x;
    if (e < E) atomicAdd(&deg[dst[e]], 1.0f);
}

__global__ __launch_bounds__(256) void dsq_kernel(const float* __restrict__ deg,
                                                  float* __restrict__ dsq, int n) {
    int i = blockIdx.x * 256 + threadIdx.x;
    if (i < n) dsq[i] = rsqrtf(fmaxf(deg[i], 1.0f));
}

// ---------------- atom encoder: h[n,d] = sum_f emb[f][idx[n,f]][d] ----------------
__global__ __launch_bounds__(128) void atom_enc(const int* __restrict__ h_idx,
                                                const float* __restrict__ emb,
                                                float* __restrict__ h) {
    int n = blockIdx.x;
    int d = threadIdx.x;
    float acc = 0.0f;
#pragma unroll
    for (int f = 0; f < 9; ++f) {
        int idx = h_idx[n * 9 + f];
        acc += emb[((size_t)f * 100 + idx) * GN_D + d];
    }
    h[(size_t)n * GN_D + d] = acc;
}

// ---------------- edge scatter: agg[dst] += X[src] * dsq[src] ----------------
// one wave per edge, float4 per lane (32 lanes * 4 = 128 channels)
__global__ __launch_bounds__(256) void prop_scatter(const int* __restrict__ src,
                                                    const int* __restrict__ dst,
                                                    const float* __restrict__ dsq,
                                                    const float* __restrict__ X,
                                                    float* __restrict__ agg, int E) {
    int e = blockIdx.x * 8 + (threadIdx.x >> 5);
    if (e >= E) return;
    int lane = threadIdx.x & 31;
    int s = src[e];
    int d = dst[e];
    float scale = dsq[s];
    const float4* xs = (const float4*)(X + (size_t)s * GN_D);
    float4 v = xs[lane];
    float* ap = agg + (size_t)d * GN_D + lane * 4;
    atomicAdd(ap + 0, v.x * scale);
    atomicAdd(ap + 1, v.y * scale);
    atomicAdd(ap + 2, v.z * scale);
    atomicAdd(ap + 3, v.w * scale);
}

// X1 = -prop(X0) = -(agg * dsq[row])       (re_norm == 1)
__global__ __launch_bounds__(256) void combine1(const float* __restrict__ agg,
                                                const float* __restrict__ dsq,
                                                float* __restrict__ X1, int total) {
    int i = blockIdx.x * 256 + threadIdx.x;
    if (i >= total) return;
    int n = i >> 7;
    X1[i] = -(agg[i] * dsq[n]);
}

// X2 = -2*prop(X1) - X0
__global__ __launch_bounds__(256) void combine2(const float* __restrict__ agg,
                                                const float* __restrict__ dsq,
                                                const float* __restrict__ X0,
                                                float* __restrict__ X2, int total) {
    int i = blockIdx.x * 256 + threadIdx.x;
    if (i >= total) return;
    int n = i >> 7;
    X2[i] = -2.0f * agg[i] * dsq[n] - X0[i];
}

// ---------------- WMMA fp32 GEMM: G[16-row tile, 128] = [X0|X1|X2]tile @ W[384,128] ----
// block: 256 threads = 8 waves; wave w computes the 16x16 tile at columns [16w,16w+16)
__global__ __launch_bounds__(256) void cheb_gemm(const float* __restrict__ X0,
                                                 const float* __restrict__ X1,
                                                 const float* __restrict__ X2,
                                                 const float* __restrict__ W,
                                                 float* __restrict__ G) {
    __shared__ float As[16 * ASTRIDE];
    const int row0 = blockIdx.x * 16;
    const int tid = threadIdx.x;

    // cooperatively stage the 16x384 A tile (Xt = [X0 | X1 | X2]) into LDS
    for (int i = tid; i < 16 * 384; i += 256) {
        int r = i / 384;
        int k = i - r * 384;
        const float* Xs = (k < 128) ? X0 : ((k < 256) ? X1 : X2);
        int kk = k & 127;
        As[r * ASTRIDE + k] = Xs[(size_t)(row0 + r) * GN_D + kk];
    }
    __syncthreads();

    const int wave = tid >> 5;
    const int lane = tid & 31;
    const int col0 = wave * 16;
    const int m = lane & 15;            // A row for this lane; also B column
    const int khalf = (lane >> 4) * 2;  // lanes 0-15 -> K{0,1}; lanes 16-31 -> K{2,3}

    v8f acc = {};
    for (int k0 = 0; k0 < 384; k0 += 4) {
        v2f a, b;
        const float* ap = &As[m * ASTRIDE + k0 + khalf];
        a.x = ap[0];
        a.y = ap[1];
        const float* wp = W + (size_t)(k0 + khalf) * GN_D + col0 + m;
        b.x = wp[0];
        b.y = wp[GN_D];
        acc = __builtin_amdgcn_wmma_f32_16x16x4_f32(false, a, false, b,
                                                    (short)0, acc, false, false);
    }

    // D layout: VGPR j -> row j (lanes 0-15) or row j+8 (lanes 16-31), col = lane&15
    const int n = lane & 15;
    const int mbase = (lane >> 4) * 8;
#pragma unroll
    for (int j = 0; j < 8; ++j) {
        G[(size_t)(row0 + mbase + j) * GN_D + col0 + n] = acc[j];
    }
}

// ---------------- batchnorm stats: one block per channel ----------------
__global__ __launch_bounds__(256) void bn_stats(const float* __restrict__ G,
                                                float* __restrict__ mu,
                                                float* __restrict__ rvar, int N) {
    int c = blockIdx.x;
    float s = 0.0f, s2 = 0.0f;
    for (int n = threadIdx.x; n < N; n += 256) {
        float v = G[(size_t)n * GN_D + c];
        s += v;
        s2 += v * v;
    }
    __shared__ float sh[256];
    __shared__ float sh2[256];
    sh[threadIdx.x] = s;
    sh2[threadIdx.x] = s2;
    __syncthreads();
    for (int off = 128; off > 0; off >>= 1) {
        if (threadIdx.x < off) {
            sh[threadIdx.x] += sh[threadIdx.x + off];
            sh2[threadIdx.x] += sh2[threadIdx.x + off];
        }
        __syncthreads();
    }
    if (threadIdx.x == 0) {
        float m = sh[0] / (float)N;
        float var = sh2[0] / (float)N - m * m;   // biased variance (torch training BN)
        mu[c] = m;
        rvar[c] = rsqrtf(var + 1e-5f);
    }
}

// h = h + relu( (G - mu)*rvar*gamma + beta )
__global__ __launch_bounds__(256) void bn_apply(float* __restrict__ h,
                                                const float* __restrict__ G,
                                                const float* __restrict__ mu,
                                                const float* __restrict__ rvar,
                                                const float* __restrict__ gamma,
                                                const float* __restrict__ beta, int total) {
    int i = blockIdx.x * 256 + threadIdx.x;
    if (i >= total) return;
    int c = i & 127;
    float v = (G[i] - mu[c]) * rvar[c] * gamma[c] + beta[c];
    h[i] += fmaxf(v, 0.0f);
}

// ---------------- mean pooling ----------------
__global__ __launch_bounds__(128) void pool_scatter(const float* __restrict__ h,
                                                    const int* __restrict__ n2g,
                                                    float* __restrict__ pool,
                                                    float* __restrict__ cnt) {
    int n = blockIdx.x;
    int d = threadIdx.x;
    int g = n2g[n];
    atomicAdd(&pool[(size_t)g * GN_D + d], h[(size_t)n * GN_D + d]);
    if (d == 0) atomicAdd(&cnt[g], 1.0f);
}

// ---------------- MLP readout: one block per graph ----------------
__global__ __launch_bounds__(128) void readout(const float* __restrict__ pool,
                                               const float* __restrict__ cnt,
                                               const float* __restrict__ W0,
                                               const float* __restrict__ b0,
                                               const float* __restrict__ W1,
                                               const float* __restrict__ b1,
                                               const float* __restrict__ W2,
                                               const float* __restrict__ b2,
                                               float* __restrict__ out) {
    __shared__ float hg[128];
    __shared__ float y0[64];
    __shared__ float y1[32];
    int g = blockIdx.x;
    int t = threadIdx.x;
    float c = fmaxf(cnt[g], 1.0f);
    hg[t] = pool[(size_t)g * GN_D + t] / c;
    __syncthreads();
    if (t < 64) {
        float a = b0[t];
        for (int k = 0; k < 128; ++k) a += hg[k] * W0[k * 64 + t];
        y0[t] = fmaxf(a, 0.0f);
    }
    __syncthreads();
    if (t < 32) {
        float a = b1[t];
        for (int k = 0; k < 64; ++k) a += y0[k] * W1[k * 32 + t];
        y1[t] = fmaxf(a, 0.0f);
    }
    __syncthreads();
    float a = b2[t];
    for (int k = 0; k < 32; ++k) a += y1[k] * W2[k * 128 + t];
    out[(size_t)g * 128 + t] = a;
}

extern "C" void kernel_launch(void* const* d_in, const int* in_sizes, int n_in,
                              void* d_out, int out_size, void* d_ws, size_t ws_size,
                              hipStream_t stream) {
    const int*   h_idx   = (const int*)d_in[0];
    // d_in[1] e_idx  : unused (bond path never feeds the output)
    const int*   src     = (const int*)d_in[2];
    const int*   dst     = (const int*)d_in[3];
    const int*   n2g     = (const int*)d_in[4];
    const float* atom_e  = (const float*)d_in[5];
    // d_in[6] bond_emb : unused
    const float* layer_W = (const float*)d_in[7];   // [4,384,128]
    const float* gamma   = (const float*)d_in[8];   // [4,128]
    const float* beta    = (const float*)d_in[9];   // [4,128]
    const float* ro_W0   = (const float*)d_in[10];
    const float* ro_b0   = (const float*)d_in[11];
    const float* ro_W1   = (const float*)d_in[12];
    const float* ro_b1   = (const float*)d_in[13];
    const float* ro_W2   = (const float*)d_in[14];
    const float* ro_b2   = (const float*)d_in[15];
    float*       out     = (float*)d_out;

    const size_t ND = (size_t)GN_N * GN_D;          // 6.4M floats
    float* ws   = (float*)d_ws;
    float* h    = ws;                                // X0 / running h
    float* X1   = h + ND;
    float* X2   = X1 + ND;
    float* agg  = X2 + ND;                           // deg, prop accumulator, GEMM out G
    float* dsq  = agg + ND;                          // [N]
    float* mu   = dsq + GN_N;                        // [128]
    float* rvar = mu + GN_D;                         // [128]
    float* pool = rvar + GN_D;                       // [B*D]
    float* cnt  = pool + (size_t)GN_B * GN_D;        // [B]

    const int ndTotal = GN_N * GN_D;
    const int gridND  = (ndTotal + 255) / 256;
    const int gridN   = (GN_N + 255) / 256;
    const int gridE   = (GN_E + 255) / 256;
    const int gridEw  = (GN_E + 7) / 8;              // 8 edges (waves) per block

    // degrees -> d^-1/2
    fillz<<<gridN, 256, 0, stream>>>(agg, GN_N);
    deg_scatter<<<gridE, 256, 0, stream>>>(dst, agg, GN_E);
    dsq_kernel<<<gridN, 256, 0, stream>>>(agg, dsq, GN_N);

    // atom encoder
    atom_enc<<<GN_N, 128, 0, stream>>>(h_idx, atom_e, h);

    for (int l = 0; l < GN_L; ++l) {
        // X1 = -prop(X0)
        fillz<<<gridND, 256, 0, stream>>>(agg, ndTotal);
        prop_scatter<<<gridEw, 256, 0, stream>>>(src, dst, dsq, h, agg, GN_E);
        combine1<<<gridND, 256, 0, stream>>>(agg, dsq, X1, ndTotal);
        // X2 = -2*prop(X1) - X0
        fillz<<<gridND, 256, 0, stream>>>(agg, ndTotal);
        prop_scatter<<<gridEw, 256, 0, stream>>>(src, dst, dsq, X1, agg, GN_E);
        combine2<<<gridND, 256, 0, stream>>>(agg, dsq, h, X2, ndTotal);
        // G = [X0|X1|X2] @ W_l   (WMMA f32, G aliases agg — safe after combine2)
        cheb_gemm<<<GN_N / 16, 256, 0, stream>>>(h, X1, X2,
                                                 layer_W + (size_t)l * 384 * GN_D, agg);
        // batchnorm + relu + residual
        bn_stats<<<GN_D, 256, 0, stream>>>(agg, mu, rvar, GN_N);
        bn_apply<<<gridND, 256, 0, stream>>>(h, agg, mu, rvar,
                                             gamma + l * GN_D, beta + l * GN_D, ndTotal);
    }

    // mean pool per graph
    fillz<<<(GN_B * GN_D + 255) / 256, 256, 0, stream>>>(pool, GN_B * GN_D);
    fillz<<<1, 256, 0, stream>>>(cnt, GN_B);
    pool_scatter<<<GN_N, 128, 0, stream>>>(h, n2g, pool, cnt);

    // MLP readout
    readout<<<GN_B, 128, 0, stream>>>(pool, cnt, ro_W0, ro_b0, ro_W1, ro_b1,
                                      ro_W2, ro_b2, out);
}